// singleLayer_51488067944601
// MI455X (gfx1250) — compile-verified
//
#include <hip/hip_runtime.h>
#include <hip/hip_bf16.h>

// N=1024, D=32, IN_F=128, OUT_F=128
// Outputs concatenated: f (N*128) | dfdx (N*32*128) | d2fd2x (N*32*32*128)
// Workspace: dsig (N*128) | d2sig (N*128) | A = dudx@K (N*32*128)  (~17.8 MB)

#define N_    1024
#define D_    32
#define INF_  128
#define OUTF_ 128

typedef __attribute__((ext_vector_type(2))) float v2f;
typedef __attribute__((ext_vector_type(8))) float v8f;

// ---------------------------------------------------------------------------
// WMMA core, V_WMMA_F32_16X16X4_F32 (K=128 -> 32 WMMAs per 16x16 tile).
// A-frag (16x4): lanes 0-15 row=lane hold (k0,k0+1); lanes 16-31 hold
// (k0+2,k0+3) -> one 8B load/lane/k-step.
// B-frag (4x16): v0 = rows k0|k0+2 striped over lanes, v1 = k0+1|k0+3.
// C/D (16x16): lane = column (lane&15), VGPR i = row i + 8*(lane>=16).
// ---------------------------------------------------------------------------

__device__ __forceinline__ void load_k_slab(const float* __restrict__ Kmat,
                                            int O0, int lane, v2f (&B)[32]) {
  const int col = O0 + (lane & 15);
  const int khi = (lane >= 16) ? 2 : 0;
#pragma unroll
  for (int kk = 0; kk < 32; ++kk) {
    const int kb = kk * 4 + khi;
    v2f b;
    b.x = Kmat[(size_t)kb * OUTF_ + col];
    b.y = Kmat[(size_t)(kb + 1) * OUTF_ + col];
    B[kk] = b;
  }
}

__device__ __forceinline__ v8f gemm_tile(const float* __restrict__ rowbase,
                                         int khi, const v2f (&B)[32]) {
  v8f acc = {};
#pragma unroll
  for (int kk = 0; kk < 32; ++kk) {
    v2f a = *(const v2f*)(rowbase + kk * 4 + khi);
    acc = __builtin_amdgcn_wmma_f32_16x16x4_f32(false, a, false, B[kk],
                                                (short)0, acc, false, false);
  }
  return acc;
}

// ---------------------------------------------------------------------------
// Kernel 1: z = u@K + b ; f = tanh(z) ; dsig = 1-f^2 ; d2sig = -2 f dsig
// ---------------------------------------------------------------------------
__global__ __launch_bounds__(256) void k1_head(
    const float* __restrict__ u, const float* __restrict__ Kmat,
    const float* __restrict__ bvec, float* __restrict__ f_out,
    float* __restrict__ dsig, float* __restrict__ d2sig, int tilesPerBlock) {
  const int lane = threadIdx.x & 31;
  const int wave = threadIdx.x >> 5;
  const int O0   = wave * 16;
  const int khi  = (lane >= 16) ? 2 : 0;
  const int c    = O0 + (lane & 15);

  v2f B[32];
  load_k_slab(Kmat, O0, lane, B);
  const float bc = bvec[c];

  for (int t = 0; t < tilesPerBlock; ++t) {
    const int T  = blockIdx.x * tilesPerBlock + t;      // row tile over N
    const int R0 = T * 16;
    const float* rowbase = u + (size_t)(R0 + (lane & 15)) * INF_;
    v8f acc = gemm_tile(rowbase, khi, B);
    const int rbase = R0 + ((lane >> 4) << 3);
#pragma unroll
    for (int i = 0; i < 8; ++i) {
      const size_t o = (size_t)(rbase + i) * OUTF_ + c;
      const float z  = acc[i] + bc;
      const float fv = tanhf(z);
      const float ds = 1.0f - fv * fv;
      f_out[o] = fv;
      dsig[o]  = ds;
      d2sig[o] = -2.0f * fv * ds;
    }
  }
}

// ---------------------------------------------------------------------------
// Kernel 2: A = dudx@K (stored to ws) ; dfdx = dsig * A
// ---------------------------------------------------------------------------
__global__ __launch_bounds__(256) void k2_grad(
    const float* __restrict__ dudx, const float* __restrict__ Kmat,
    const float* __restrict__ dsig, float* __restrict__ A_ws,
    float* __restrict__ dfdx_out, int tilesPerBlock) {
  const int lane = threadIdx.x & 31;
  const int wave = threadIdx.x >> 5;
  const int O0   = wave * 16;
  const int khi  = (lane >= 16) ? 2 : 0;
  const int c    = O0 + (lane & 15);

  v2f B[32];
  load_k_slab(Kmat, O0, lane, B);

  for (int t = 0; t < tilesPerBlock; ++t) {
    const int T  = blockIdx.x * tilesPerBlock + t;      // tile over N*D rows
    const int R0 = T * 16;                              // rows (n*32 + d)
    const int n  = R0 >> 5;                             // tile stays in one n
    const float* rowbase = dudx + (size_t)(R0 + (lane & 15)) * INF_;
    v8f acc = gemm_tile(rowbase, khi, B);
    const float ds = dsig[(size_t)n * OUTF_ + c];
    const int rbase = R0 + ((lane >> 4) << 3);
#pragma unroll
    for (int i = 0; i < 8; ++i) {
      const size_t o = (size_t)(rbase + i) * OUTF_ + c;
      const float a  = acc[i];
      A_ws[o]     = a;
      dfdx_out[o] = ds * a;
    }
  }
}

// ---------------------------------------------------------------------------
// Kernel 3 (dominant): B = d2ud2x@K, fused epilogue
//   d2fd2x[n,d,e,o] = dsig[n,o]*B[...] + d2sig[n,o]*A[n,d,o]*A[n,e,o]
//
// Wide tiles to cut L0 operand amplification: each wave computes 16x32
// (two o-slabs, two K-slabs in 128 VGPRs, one A-frag load feeds 2 WMMAs).
// Waves 0-3 and 4-7 take different row-tiles, so each 16x128 input tile is
// read by 4 waves (was 8): ~2 GB of L0 reads over the whole kernel vs the
// 512 MB HBM stream -> well under WGP$ bandwidth at the 46 us HBM floor.
// ---------------------------------------------------------------------------
__global__ __launch_bounds__(256) void k3_hess(
    const float* __restrict__ d2ud2x, const float* __restrict__ Kmat,
    const float* __restrict__ dsig, const float* __restrict__ d2sig,
    const float* __restrict__ A_ws, float* __restrict__ out3,
    int pairsPerBlock) {
  const int lane    = threadIdx.x & 31;
  const int wave    = threadIdx.x >> 5;
  const int O0      = (wave & 3) * 32;        // 32-wide column slab
  const int rowHalf = wave >> 2;              // which row-tile of the pair
  const int khi     = (lane >= 16) ? 2 : 0;
  const int c0      = O0 + (lane & 15);
  const int c1      = c0 + 16;

  v2f B0[32], B1[32];
  load_k_slab(Kmat, O0,      lane, B0);
  load_k_slab(Kmat, O0 + 16, lane, B1);

  for (int t = 0; t < pairsPerBlock; ++t) {
    const int T   = (blockIdx.x * pairsPerBlock + t) * 2 + rowHalf; // [0,65536)
    const int n   = T >> 6;                             // 64 tiles per n
    const int rem = T & 63;
    const int d   = rem >> 1;
    const int e0  = (rem & 1) << 4;
    const int R0  = ((n * D_ + d) * D_ + e0);           // flat row base
    const float* rowbase = d2ud2x + (size_t)(R0 + (lane & 15)) * INF_;

    if (t + 1 < pairsPerBlock)                          // stream-ahead hint
      __builtin_prefetch((const void*)(rowbase + 32 * INF_), 0, 0);

    v8f acc0 = {};
    v8f acc1 = {};
#pragma unroll
    for (int kk = 0; kk < 32; ++kk) {
      v2f a = *(const v2f*)(rowbase + kk * 4 + khi);
      acc0 = __builtin_amdgcn_wmma_f32_16x16x4_f32(false, a, false, B0[kk],
                                                   (short)0, acc0, false, false);
      acc1 = __builtin_amdgcn_wmma_f32_16x16x4_f32(false, a, false, B1[kk],
                                                   (short)0, acc1, false, false);
    }

    const size_t nd  = (size_t)(n * D_ + d) * OUTF_;
    const size_t nb  = (size_t)n * OUTF_;
    const float ds0  = dsig[nb + c0],  ds1  = dsig[nb + c1];
    const float d2s0 = d2sig[nb + c0], d2s1 = d2sig[nb + c1];
    const float Ad0  = A_ws[nd + c0],  Ad1  = A_ws[nd + c1];
    const int rlo    = ((lane >> 4) << 3);
#pragma unroll
    for (int i = 0; i < 8; ++i) {
      const int e       = e0 + rlo + i;
      const size_t ne   = (size_t)(n * D_ + e) * OUTF_;
      const size_t obase = (size_t)(R0 + rlo + i) * OUTF_;
      out3[obase + c0] = ds0 * acc0[i] + d2s0 * Ad0 * A_ws[ne + c0];
      out3[obase + c1] = ds1 * acc1[i] + d2s1 * Ad1 * A_ws[ne + c1];
    }
  }
}

// ---------------------------------------------------------------------------
extern "C" void kernel_launch(void* const* d_in, const int* in_sizes, int n_in,
                              void* d_out, int out_size, void* d_ws, size_t ws_size,
                              hipStream_t stream) {
  const float* u      = (const float*)d_in[0];
  const float* dudx   = (const float*)d_in[1];
  const float* d2ud2x = (const float*)d_in[2];
  const float* Kmat   = (const float*)d_in[3];
  const float* bvec   = (const float*)d_in[4];

  float* out      = (float*)d_out;
  float* f_out    = out;
  float* dfdx_out = out + (size_t)N_ * OUTF_;
  float* d2_out   = dfdx_out + (size_t)N_ * D_ * OUTF_;

  float* ws    = (float*)d_ws;
  float* dsig  = ws;
  float* d2sig = ws + (size_t)N_ * OUTF_;
  float* A_ws  = d2sig + (size_t)N_ * OUTF_;

  // K1: 64 row tiles  -> 8 blocks x 8 tiles
  k1_head<<<8, 256, 0, stream>>>(u, Kmat, bvec, f_out, dsig, d2sig, 8);
  // K2: 2048 row tiles -> 256 blocks x 8 tiles
  k2_grad<<<256, 256, 0, stream>>>(dudx, Kmat, dsig, A_ws, dfdx_out, 8);
  // K3: 65536 row tiles = 32768 pairs -> 2048 blocks x 16 pairs
  k3_hess<<<2048, 256, 0, stream>>>(d2ud2x, Kmat, dsig, d2sig, A_ws, d2_out, 16);
}